// LGDE_32263794328249
// MI455X (gfx1250) — compile-verified
//
#include <hip/hip_runtime.h>
#include <math.h>

// ---------------- problem constants ----------------
constexpr int kB   = 16;
constexpr int kT   = 8192;
constexpr int kCIN = 128;
constexpr int kD   = 32;    // head dim (C_OUT)
constexpr int kH   = 4;
constexpr int kM   = 64;    // random features
constexpr float kScale = 0.4204482076268573f;   // 32^-0.25  (TAU=1 -> inv_sqrt_tau=1)
constexpr float kRatio = 0.125f;                // 1/sqrt(M)
constexpr float kEps   = 1e-6f;

typedef __attribute__((ext_vector_type(2))) float v2f;
typedef __attribute__((ext_vector_type(8))) float v8f;

__device__ __forceinline__ v8f wmma4(v2f a, v2f b, v8f c) {
  // D = A(16x4,f32) x B(4x16,f32) + C(16x16,f32)
  return __builtin_amdgcn_wmma_f32_16x16x4_f32(false, a, false, b, (short)0, c,
                                               false, false);
}

// monotonic float<->uint encoding for atomicMax
__device__ __forceinline__ unsigned int enc_f32(float f) {
  unsigned int u = __float_as_uint(f);
  return (u & 0x80000000u) ? ~u : (u | 0x80000000u);
}
__device__ __forceinline__ float dec_f32(unsigned int u) {
  return __uint_as_float((u & 0x80000000u) ? (u ^ 0x80000000u) : ~u);
}

// =====================================================================
// Phase 1: global max over (t,m) of key dash, per (b,h).
// grid: (16 chunks, H, B), block: 128 (4 waves x 16 rows)
// =====================================================================
__global__ void __launch_bounds__(128)
perf_kmax_kernel(const float* __restrict__ x,
                 const float* __restrict__ Wk_w,
                 const float* __restrict__ Wk_b,
                 const float* __restrict__ proj,
                 unsigned int* __restrict__ mx_enc)
{
  __shared__ float khs[64][33];
  __shared__ float red[128];

  const int chunk = blockIdx.x, h = blockIdx.y, b = blockIdx.z;
  const int tid  = threadIdx.x;
  const int lane = tid & 31, wave = tid >> 5;
  const int lm   = lane & 15, half = lane >> 4;
  const int rows0 = wave * 16;

  float vmax = -3.0e38f;

  for (int sc = 0; sc < 8; ++sc) {
    const int t0 = chunk * 512 + sc * 64;
    const float* xrow = x + ((size_t)b * kT + t0 + rows0 + lm) * kCIN;

    // k head-slice GEMM: [16x128] @ [128x32] -> khs strip
    for (int ct = 0; ct < 2; ++ct) {
      v8f c = {};
      const float* wk = Wk_w + (size_t)(h * kD + ct * 16 + lm) * kCIN;
      for (int ks = 0; ks < 32; ++ks) {
        const int k = ks * 4 + 2 * half;
        v2f a; a.x = xrow[k]; a.y = xrow[k + 1];
        v2f bb; bb.x = wk[k]; bb.y = wk[k + 1];
        c = wmma4(a, bb, c);
      }
      const float bias = Wk_b[h * kD + ct * 16 + lm];
      for (int r = 0; r < 8; ++r)
        khs[rows0 + r + 8 * half][ct * 16 + lm] = c[r] + bias;
    }
    __syncthreads();

    // dash GEMM: [16x32] @ [32x64]; only the max is needed
    for (int mt = 0; mt < 4; ++mt) {
      v8f c = {};
      for (int ks = 0; ks < 8; ++ks) {
        const int k = ks * 4 + 2 * half;
        v2f a; a.x = khs[rows0 + lm][k]; a.y = khs[rows0 + lm][k + 1];
        v2f bb;
        bb.x = kScale * proj[(mt * 16 + lm) * kD + k];
        bb.y = kScale * proj[(mt * 16 + lm) * kD + k + 1];
        c = wmma4(a, bb, c);
      }
      for (int r = 0; r < 8; ++r) vmax = fmaxf(vmax, c[r]);
    }
    __syncthreads();
  }

  red[tid] = vmax;
  __syncthreads();
  if (tid == 0) {
    float m = red[0];
    for (int i = 1; i < 128; ++i) m = fmaxf(m, red[i]);
    atomicMax(&mx_enc[b * kH + h], enc_f32(m));
  }
}

// =====================================================================
// Phase 2: per (b,h): kf features, kvs = kf^T @ v (persistent C-frags),
// ksum = sum_t kf.  grid: (H, B), block: 128 (4 waves x 16 rows), 128 chunks.
// =====================================================================
__global__ void __launch_bounds__(128)
perf_kv_kernel(const float* __restrict__ x,
               const float* __restrict__ Wk_w, const float* __restrict__ Wk_b,
               const float* __restrict__ Wv_w, const float* __restrict__ Wv_b,
               const float* __restrict__ proj,
               const unsigned int* __restrict__ mx_enc,
               float* __restrict__ kvs_g, float* __restrict__ ksum_g)
{
  __shared__ float khs[64][33];
  __shared__ float vhs[64][33];
  __shared__ float kfs[64][65];
  __shared__ float diags[64];

  const int h = blockIdx.x, b = blockIdx.y;
  const int bh = b * kH + h;
  const int tid  = threadIdx.x;
  const int lane = tid & 31, wave = tid >> 5;
  const int lm   = lane & 15, half = lane >> 4;
  const int rows0 = wave * 16;

  const float mxk = dec_f32(mx_enc[bh]);
  const float s2  = kScale * kScale;

  v8f ckv[2] = {v8f{}, v8f{}};   // wave owns mTile=wave, dTiles 0..1
  float ksacc = 0.f;

  for (int chunk = 0; chunk < 128; ++chunk) {
    const int t0 = chunk * 64;
    const float* xrow = x + ((size_t)b * kT + t0 + rows0 + lm) * kCIN;

    // k and v head-slices share the A fragments of x
    for (int ct = 0; ct < 2; ++ct) {
      v8f ck = {}, cv = {};
      const float* wk = Wk_w + (size_t)(h * kD + ct * 16 + lm) * kCIN;
      const float* wv = Wv_w + (size_t)(h * kD + ct * 16 + lm) * kCIN;
      for (int ks = 0; ks < 32; ++ks) {
        const int k = ks * 4 + 2 * half;
        v2f a; a.x = xrow[k]; a.y = xrow[k + 1];
        v2f bk; bk.x = wk[k]; bk.y = wk[k + 1];
        v2f bv; bv.x = wv[k]; bv.y = wv[k + 1];
        ck = wmma4(a, bk, ck);
        cv = wmma4(a, bv, cv);
      }
      const float biask = Wk_b[h * kD + ct * 16 + lm];
      const float biasv = Wv_b[h * kD + ct * 16 + lm];
      for (int r = 0; r < 8; ++r) {
        const int row = rows0 + r + 8 * half;
        khs[row][ct * 16 + lm] = ck[r] + biask;
        vhs[row][ct * 16 + lm] = cv[r] + biasv;
      }
    }
    __syncthreads();

    if (tid < 64) {
      float su = 0.f;
      for (int d = 0; d < kD; ++d) { const float kv = khs[tid][d]; su += kv * kv; }
      diags[tid] = 0.5f * s2 * su;
    }
    __syncthreads();

    // dash -> kf (transform at C-store)
    for (int mt = 0; mt < 4; ++mt) {
      v8f c = {};
      for (int ks = 0; ks < 8; ++ks) {
        const int k = ks * 4 + 2 * half;
        v2f a; a.x = khs[rows0 + lm][k]; a.y = khs[rows0 + lm][k + 1];
        v2f bb;
        bb.x = kScale * proj[(mt * 16 + lm) * kD + k];
        bb.y = kScale * proj[(mt * 16 + lm) * kD + k + 1];
        c = wmma4(a, bb, c);
      }
      for (int r = 0; r < 8; ++r) {
        const int row = rows0 + r + 8 * half;
        kfs[row][mt * 16 + lm] =
            kRatio * (__expf(c[r] - diags[row] - mxk) + kEps);
      }
    }
    __syncthreads();

    if (tid < 64) {
      float a = 0.f;
      for (int t = 0; t < 64; ++t) a += kfs[t][tid];
      ksacc += a;
    }

    // kvs accumulate: A = kf^T [M=wave tile][K=t], B = vh
    for (int dt = 0; dt < 2; ++dt) {
      for (int ks = 0; ks < 16; ++ks) {
        const int k = ks * 4 + 2 * half;
        v2f a; a.x = kfs[k][wave * 16 + lm]; a.y = kfs[k + 1][wave * 16 + lm];
        v2f bb; bb.x = vhs[k][dt * 16 + lm]; bb.y = vhs[k + 1][dt * 16 + lm];
        ckv[dt] = wmma4(a, bb, ckv[dt]);
      }
    }
    __syncthreads();
  }

  // store kvs[bh][m][d]
  for (int dt = 0; dt < 2; ++dt)
    for (int r = 0; r < 8; ++r) {
      const int m = wave * 16 + r + 8 * half;
      const int d = dt * 16 + lm;
      kvs_g[(size_t)bh * (kM * kD) + m * kD + d] = ckv[dt][r];
    }
  if (tid < 64) ksum_g[bh * kM + tid] = ksacc;
}

// =====================================================================
// Phase 3: q features, num/den, output projection (partial per-head
// accumulation into persistent C-frags).
// grid: (T/64, B), block: 128 (4 waves x 16 rows)
// =====================================================================
__global__ void __launch_bounds__(128)
perf_qout_kernel(const float* __restrict__ x,
                 const float* __restrict__ Wq_w, const float* __restrict__ Wq_b,
                 const float* __restrict__ Wo_w, const float* __restrict__ Wo_b,
                 const float* __restrict__ proj,
                 const float* __restrict__ kvs_g, const float* __restrict__ ksum_g,
                 float* __restrict__ out)
{
  __shared__ float qs[64][129];
  __shared__ float dashs[64][65];
  __shared__ float atth[64][33];
  __shared__ float diags[64];
  __shared__ float dens[64];

  const int tc = blockIdx.x, b = blockIdx.y;
  const int t0 = tc * 64;
  const int tid  = threadIdx.x;
  const int lane = tid & 31, wave = tid >> 5;
  const int lm   = lane & 15, half = lane >> 4;
  const int rows0 = wave * 16;
  const float s2 = kScale * kScale;

  // 1) q GEMM: [64x128] @ [128x128]
  const float* xrow = x + ((size_t)b * kT + t0 + rows0 + lm) * kCIN;
  for (int ct = 0; ct < 8; ++ct) {
    v8f c = {};
    const float* wq = Wq_w + (size_t)(ct * 16 + lm) * kCIN;
    for (int ks = 0; ks < 32; ++ks) {
      const int k = ks * 4 + 2 * half;
      v2f a; a.x = xrow[k]; a.y = xrow[k + 1];
      v2f bb; bb.x = wq[k]; bb.y = wq[k + 1];
      c = wmma4(a, bb, c);
    }
    const float bias = Wq_b[ct * 16 + lm];
    for (int r = 0; r < 8; ++r)
      qs[rows0 + r + 8 * half][ct * 16 + lm] = c[r] + bias;
  }
  __syncthreads();

  v8f oc[2] = {v8f{}, v8f{}};    // output C-frags, accumulated over heads

  for (int h = 0; h < kH; ++h) {
    const int bh = b * kH + h;

    if (tid < 64) {
      float su = 0.f;
      for (int d = 0; d < kD; ++d) {
        const float qv = qs[tid][h * kD + d];
        su += qv * qv;
      }
      diags[tid] = 0.5f * s2 * su;
    }
    __syncthreads();

    // dash GEMM: [64x32] @ [32x64]
    for (int mt = 0; mt < 4; ++mt) {
      v8f c = {};
      for (int ks = 0; ks < 8; ++ks) {
        const int k = ks * 4 + 2 * half;
        v2f a; a.x = qs[rows0 + lm][h * kD + k]; a.y = qs[rows0 + lm][h * kD + k + 1];
        v2f bb;
        bb.x = kScale * proj[(mt * 16 + lm) * kD + k];
        bb.y = kScale * proj[(mt * 16 + lm) * kD + k + 1];
        c = wmma4(a, bb, c);
      }
      for (int r = 0; r < 8; ++r)
        dashs[rows0 + r + 8 * half][mt * 16 + lm] = c[r];
    }
    __syncthreads();

    // qf + den (per-row)
    if (tid < 64) {
      float mx = -3.0e38f;
      for (int m = 0; m < kM; ++m) mx = fmaxf(mx, dashs[tid][m]);
      const float* ksum = ksum_g + bh * kM;
      float den = 0.f;
      const float dg = diags[tid];
      for (int m = 0; m < kM; ++m) {
        const float v = kRatio * (__expf(dashs[tid][m] - dg - mx) + kEps);
        dashs[tid][m] = v;
        den += v * ksum[m];
      }
      dens[tid] = den;
    }
    __syncthreads();

    // num GEMM: [64x64] @ [64x32]; divide by den at store
    const float* kv = kvs_g + (size_t)bh * (kM * kD);
    for (int dt = 0; dt < 2; ++dt) {
      v8f c = {};
      for (int ks = 0; ks < 16; ++ks) {
        const int k = ks * 4 + 2 * half;
        v2f a; a.x = dashs[rows0 + lm][k]; a.y = dashs[rows0 + lm][k + 1];
        v2f bb; bb.x = kv[k * kD + dt * 16 + lm]; bb.y = kv[(k + 1) * kD + dt * 16 + lm];
        c = wmma4(a, bb, c);
      }
      for (int r = 0; r < 8; ++r) {
        const int row = rows0 + r + 8 * half;
        atth[row][dt * 16 + lm] = c[r] / dens[row];
      }
    }
    __syncthreads();

    // partial output GEMM: att_h [64x32] @ Wo_h^T [32x32], accumulate
    for (int ot = 0; ot < 2; ++ot) {
      for (int ks = 0; ks < 8; ++ks) {
        const int k = ks * 4 + 2 * half;
        v2f a; a.x = atth[rows0 + lm][k]; a.y = atth[rows0 + lm][k + 1];
        const float* wo = Wo_w + (size_t)(ot * 16 + lm) * (kH * kD) + h * kD;
        v2f bb; bb.x = wo[k]; bb.y = wo[k + 1];
        oc[ot] = wmma4(a, bb, oc[ot]);
      }
    }
    __syncthreads();
  }

  // store output + bias: out[b][t][co], co in 0..31
  for (int ot = 0; ot < 2; ++ot) {
    const float bias = Wo_b[ot * 16 + lm];
    for (int r = 0; r < 8; ++r) {
      const int row = rows0 + r + 8 * half;
      out[((size_t)b * kT + t0 + row) * kD + ot * 16 + lm] = oc[ot][r] + bias;
    }
  }
}

// =====================================================================
// launch
// =====================================================================
extern "C" void kernel_launch(void* const* d_in, const int* in_sizes, int n_in,
                              void* d_out, int out_size, void* d_ws, size_t ws_size,
                              hipStream_t stream) {
  (void)in_sizes; (void)n_in; (void)out_size; (void)ws_size;
  const float* x    = (const float*)d_in[0];
  const float* Wq_w = (const float*)d_in[1];
  const float* Wq_b = (const float*)d_in[2];
  const float* Wk_w = (const float*)d_in[3];
  const float* Wk_b = (const float*)d_in[4];
  const float* Wv_w = (const float*)d_in[5];
  const float* Wv_b = (const float*)d_in[6];
  const float* Wo_w = (const float*)d_in[7];
  const float* Wo_b = (const float*)d_in[8];
  const float* proj = (const float*)d_in[9];

  unsigned int* mx_enc = (unsigned int*)d_ws;                      // 64 * 4B
  float* ksum = (float*)((char*)d_ws + 256);                       // 4096 * 4B
  float* kvs  = (float*)((char*)d_ws + 256 + 16384);               // 131072 * 4B

  // re-init only the atomic-max region every call (graph-capturable)
  hipMemsetAsync(d_ws, 0, 256, stream);

  perf_kmax_kernel<<<dim3(16, kH, kB), 128, 0, stream>>>(x, Wk_w, Wk_b, proj, mx_enc);
  perf_kv_kernel<<<dim3(kH, kB), 128, 0, stream>>>(x, Wk_w, Wk_b, Wv_w, Wv_b, proj,
                                                   mx_enc, kvs, ksum);
  perf_qout_kernel<<<dim3(kT / 64, kB), 128, 0, stream>>>(x, Wq_w, Wq_b, Wo_w, Wo_b,
                                                          proj, kvs, ksum,
                                                          (float*)d_out);
}